// DecoupleImage_24902220382275
// MI455X (gfx1250) — compile-verified
//
#include <hip/hip_runtime.h>
#include <math.h>

// ---------------------------------------------------------------------------
// L0-smoothing image decoupling on gfx1250 (MI455X).
//  - 14 lambda iterations, each: fused stencil/threshold kernel -> row FFT ->
//    column FFT + 1/Denormin solve + inverse column FFT (fused) -> inverse row FFT.
//  - 512-pt FFTs built from DFT-16 matmuls on V_WMMA_F32_16X16X4_F32 (fp32 tensor
//    core path; fp32 is mandatory for FFT accuracy on this workload).
//  - All buffers L2-resident (48MB ws, 192MB L2); every global access coalesced
//    via LDS transposes. wave32: one FFT per wave, 16 waves per workgroup.
// ---------------------------------------------------------------------------

typedef float v2f __attribute__((ext_vector_type(2)));
typedef float v8f __attribute__((ext_vector_type(8)));

#define ALPHA_C 0.8f
#define BETA_C  0.05f
#define HWSZ    (512 * 512)

__device__ __forceinline__ v8f wmma4(v2f a, v2f b, v8f c) {
  // D = A(16x4,f32) x B(4x16,f32) + C(16x16,f32)
  return __builtin_amdgcn_wmma_f32_16x16x4_f32(false, a, false, b, (short)0, c,
                                               false, false);
}

// One 512-point complex FFT per wave, data in this wave's LDS slice re[0..511],
// im[0..511], natural order in and out. sgn=+1 forward (e^{-i}), -1 inverse
// (unscaled). Decomposition: 512 = 16 x 32, DFT16 stages as WMMA matmuls,
// DFT32 = radix-2 + DFT16.
__device__ __forceinline__ void wave_fft512(float* re, float* im, float sgn,
                                            bool realIn) {
  const int lane = threadIdx.x & 31;
  const int hi   = lane >> 4;   // which half-wave
  const int mr   = lane & 15;   // M (A/C rows) or N (B/C cols)
  const float PI = 3.14159265358979323846f;

  // DFT16 constants in WMMA A-operand layout, 4 K-chunks of 16x4.
  // A(M,K) = W16^{M*K}, lane: M = mr, K = 4q + 2*hi + p.
  v2f ar[4], ai[4], ain[4];
#pragma unroll
  for (int q = 0; q < 4; ++q)
#pragma unroll
    for (int p = 0; p < 2; ++p) {
      int K  = 4 * q + 2 * hi + p;
      int mk = (mr * K) & 15;  // W16 periodic
      float s, c;
      __sincosf(-sgn * (PI / 8.0f) * (float)mk, &s, &c);
      ar[q][p] = c; ai[q][p] = s; ain[q][p] = -s;
    }

  // ---- Stage A: B1[k1][n2] = sum_n1 W16^{k1*n1} x[n1*32+n2], 2 tiles in n2.
  v8f cr[2], ci[2];
#pragma unroll
  for (int t = 0; t < 2; ++t) {
    v2f br[4], bi[4];
#pragma unroll
    for (int q = 0; q < 4; ++q)
#pragma unroll
      for (int p = 0; p < 2; ++p) {
        int n1 = 4 * q + 2 * hi + p;          // B-operand K index
        int idx = n1 * 32 + 16 * t + mr;      // x[n1*32 + n2]
        br[q][p] = re[idx];
        bi[q][p] = realIn ? 0.0f : im[idx];
      }
    v8f aRe = {}, aIm = {};
#pragma unroll
    for (int q = 0; q < 4; ++q) aRe = wmma4(ar[q], br[q], aRe);
#pragma unroll
    for (int q = 0; q < 4; ++q) aIm = wmma4(ai[q], br[q], aIm);
    if (!realIn) {
#pragma unroll
      for (int q = 0; q < 4; ++q) aRe = wmma4(ain[q], bi[q], aRe);
#pragma unroll
      for (int q = 0; q < 4; ++q) aIm = wmma4(ar[q], bi[q], aIm);
    }
    cr[t] = aRe; ci[t] = aIm;
  }

  // ---- Twiddle W512^{k1*n2} + radix-2 over n2 halves; stage U,V to LDS.
  // C/D layout: vgpr v, lane -> (k1 = v + 8*hi, n2 = 16*t + mr).
#pragma unroll
  for (int v = 0; v < 8; ++v) {
    int k1 = v + 8 * hi;
    float t0r, t0i, t1r, t1i, s, c;
    __sincosf(-sgn * (2.0f * PI / 512.0f) * (float)((k1 * mr) & 511), &s, &c);
    t0r = cr[0][v] * c - ci[0][v] * s;
    t0i = cr[0][v] * s + ci[0][v] * c;
    __sincosf(-sgn * (2.0f * PI / 512.0f) * (float)((k1 * (mr + 16)) & 511), &s, &c);
    t1r = cr[1][v] * c - ci[1][v] * s;
    t1i = cr[1][v] * s + ci[1][v] * c;
    float ur = t0r + t1r, ui = t0i + t1i;
    float dr = t0r - t1r, di = t0i - t1i;
    __sincosf(-sgn * (2.0f * PI / 32.0f) * (float)mr, &s, &c);  // W32^{b}
    float vr = dr * c - di * s, vi = dr * s + di * c;
    re[k1 * 16 + mr]       = ur;  im[k1 * 16 + mr]       = ui;
    re[256 + k1 * 16 + mr] = vr;  im[256 + k1 * 16 + mr] = vi;
  }

  // ---- Stage C: R_m[d][k1] = sum_b W16^{d*b} U_m[k1][b]  (U^T as B-operand).
  v8f rR[2], rI[2];
#pragma unroll
  for (int m = 0; m < 2; ++m) {
    v2f br[4], bi[4];
#pragma unroll
    for (int q = 0; q < 4; ++q)
#pragma unroll
      for (int p = 0; p < 2; ++p) {
        int b   = 4 * q + 2 * hi + p;       // B-operand K index
        int idx = m * 256 + mr * 16 + b;    // U[k1=N][b=K]
        br[q][p] = re[idx];
        bi[q][p] = im[idx];
      }
    v8f aRe = {}, aIm = {};
#pragma unroll
    for (int q = 0; q < 4; ++q) aRe = wmma4(ar[q], br[q], aRe);
#pragma unroll
    for (int q = 0; q < 4; ++q) aRe = wmma4(ain[q], bi[q], aRe);
#pragma unroll
    for (int q = 0; q < 4; ++q) aIm = wmma4(ai[q], br[q], aIm);
#pragma unroll
    for (int q = 0; q < 4; ++q) aIm = wmma4(ar[q], bi[q], aIm);
    rR[m] = aRe; rI[m] = aIm;
  }

  // ---- Output X[(2d+m)*16 + k1] (natural order), d = v + 8*hi, k1 = mr.
#pragma unroll
  for (int m = 0; m < 2; ++m)
#pragma unroll
    for (int v = 0; v < 8; ++v) {
      int d = v + 8 * hi;
      int k = (2 * d + m) * 16 + mr;
      re[k] = rR[m][v];
      im[k] = rI[m][v];
    }
}

// ---------------- Row forward FFT (real in) -> transposed complex out -------
__global__ void __launch_bounds__(512)
k_row_fwd(const float* __restrict__ rhs, float* __restrict__ tre,
          float* __restrict__ tim) {
  __shared__ float lre[16 * 512];
  __shared__ float lim[16 * 512];
  int img = blockIdx.x >> 5;
  int h0  = (blockIdx.x & 31) * 16;
  int wv  = threadIdx.x >> 5;
  int ln  = threadIdx.x & 31;
  float* re = lre + wv * 512;
  float* im = lim + wv * 512;
  const float* src = rhs + img * HWSZ + (h0 + wv) * 512;
  for (int i = ln; i < 512; i += 32) re[i] = src[i];
  wave_fft512(re, im, 1.0f, true);
  __syncthreads();
  // T1[w][h] = X_h[w]  (coalesced 64B segments over h)
  int c = threadIdx.x & 15;
  int r = threadIdx.x >> 4;
  float* dR = tre + img * HWSZ;
  float* dI = tim + img * HWSZ;
#pragma unroll
  for (int wb = 0; wb < 16; ++wb) {
    int w = wb * 32 + r;
    dR[w * 512 + h0 + c] = lre[c * 512 + w];
    dI[w * 512 + h0 + c] = lim[c * 512 + w];
  }
}

// ------- Column FFT -> 1/(Denormin*N^2) -> inverse column FFT (fused) -------
__global__ void __launch_bounds__(512)
k_col_solve(const float* __restrict__ t1r, const float* __restrict__ t1i,
            float* __restrict__ t2r, float* __restrict__ t2i, float lam) {
  __shared__ float lre[16 * 512];
  __shared__ float lim[16 * 512];
  int img = blockIdx.x >> 5;
  int w0  = (blockIdx.x & 31) * 16;
  int wv  = threadIdx.x >> 5;
  int ln  = threadIdx.x & 31;
  int w   = w0 + wv;
  float* re = lre + wv * 512;
  float* im = lim + wv * 512;
  const float* sR = t1r + img * HWSZ + w * 512;
  const float* sI = t1i + img * HWSZ + w * 512;
  for (int i = ln; i < 512; i += 32) { re[i] = sR[i]; im[i] = sI[i]; }
  wave_fft512(re, im, 1.0f, false);
  // Denormin(h,w) = 1 + ALPHA*0.25*(aw+ah) + lam*0.0625*(aw^2+ah^2+ap^2+am^2)
  const float PI2 = 6.28318530717958647692f;
  float thw = PI2 * (float)w / 512.0f;
  float aw  = 2.0f - 2.0f * __cosf(thw);
  for (int i = ln; i < 512; i += 32) {
    float thh = PI2 * (float)i / 512.0f;
    float ah = 2.0f - 2.0f * __cosf(thh);
    float ap = 2.0f - 2.0f * __cosf(thw + thh);
    float am = 2.0f - 2.0f * __cosf(thw - thh);
    float den = 1.0f + ALPHA_C * 0.25f * (aw + ah)
              + lam * 0.0625f * (aw * aw + ah * ah + ap * ap + am * am);
    float sc = 1.0f / (den * 262144.0f);  // folds ifft2 1/(H*W)
    re[i] *= sc; im[i] *= sc;
  }
  wave_fft512(re, im, -1.0f, false);
  __syncthreads();
  // T2[h][w] (back to h-major), coalesced over w
  int c = threadIdx.x & 15;
  int r = threadIdx.x >> 4;
  float* dR = t2r + img * HWSZ;
  float* dI = t2i + img * HWSZ;
#pragma unroll
  for (int hb = 0; hb < 16; ++hb) {
    int h = hb * 32 + r;
    dR[h * 512 + w0 + c] = lre[c * 512 + h];
    dI[h * 512 + w0 + c] = lim[c * 512 + h];
  }
}

// ---------------- Inverse row FFT -> real S ---------------------------------
__global__ void __launch_bounds__(512)
k_row_inv(const float* __restrict__ t2r, const float* __restrict__ t2i,
          float* __restrict__ S) {
  __shared__ float lre[16 * 512];
  __shared__ float lim[16 * 512];
  int img = blockIdx.x >> 5;
  int h0  = (blockIdx.x & 31) * 16;
  int wv  = threadIdx.x >> 5;
  int ln  = threadIdx.x & 31;
  int h   = h0 + wv;
  float* re = lre + wv * 512;
  float* im = lim + wv * 512;
  const float* sR = t2r + img * HWSZ + h * 512;
  const float* sI = t2i + img * HWSZ + h * 512;
  for (int i = ln; i < 512; i += 32) { re[i] = sR[i]; im[i] = sI[i]; }
  wave_fft512(re, im, -1.0f, false);
  float* dst = S + img * HWSZ + h * 512;
  for (int i = ln; i < 512; i += 32) dst[i] = re[i];
}

// ------ Fused gradients + cross-channel threshold + RHS stencil kernel ------
// RHS = S0 + ALPHA*(-(lapx+lapy)(S-S0)) + lam*(lapx gxx' + lapy gyy' + lapu guu' + lapv gvv')
__global__ void __launch_bounds__(256)
k_grad_rhs(const float* __restrict__ S, const float* __restrict__ S0,
           float* __restrict__ rhs, float lam) {
  __shared__ float sS[3][20][20];      // S tile, halo 2
  __shared__ float sD[3][18][18];      // S - S0, halo 1
  __shared__ float sG[4][3][18][18];   // thresholded gxx,gyy,guu,gvv, halo 1
  int b   = blockIdx.z;
  int ty0 = blockIdx.y * 16, tx0 = blockIdx.x * 16;
  int tid = threadIdx.y * 16 + threadIdx.x;

  for (int c = 0; c < 3; ++c) {
    int base = (b * 3 + c) * HWSZ;
    for (int i = tid; i < 400; i += 256) {
      int yy = (ty0 - 2 + i / 20) & 511, xx = (tx0 - 2 + i % 20) & 511;
      sS[c][i / 20][i % 20] = S[base + yy * 512 + xx];
    }
  }
  __syncthreads();

  float thr = BETA_C / lam;
  for (int i = tid; i < 324; i += 256) {
    int j = i / 18, k = i % 18;
    int y = j + 1, x = k + 1;   // center in sS coords
    float g[3][4];
    float sum = 0.0f;
    for (int c = 0; c < 3; ++c) {
      float cen = sS[c][y][x];
      float gxx = (sS[c][y][x - 1] - 2.0f * cen + sS[c][y][x + 1]) * 0.25f;
      float gyy = (sS[c][y - 1][x] - 2.0f * cen + sS[c][y + 1][x]) * 0.25f;
      float guu = (sS[c][y - 1][x - 1] - 2.0f * cen + sS[c][y + 1][x + 1]) * 0.25f;
      float gvv = (sS[c][y - 1][x + 1] - 2.0f * cen + sS[c][y + 1][x - 1]) * 0.25f;
      sum += gxx * gxx + gyy * gyy + guu * guu + gvv * gvv;
      g[c][0] = gxx; g[c][1] = gyy; g[c][2] = guu; g[c][3] = gvv;
    }
    float f = (sum >= thr) ? 1.0f : 0.0f;
    int yy = (ty0 - 1 + j) & 511, xx = (tx0 - 1 + k) & 511;
    for (int c = 0; c < 3; ++c) {
      sG[0][c][j][k] = g[c][0] * f;
      sG[1][c][j][k] = g[c][1] * f;
      sG[2][c][j][k] = g[c][2] * f;
      sG[3][c][j][k] = g[c][3] * f;
      sD[c][j][k] = sS[c][y][x] - S0[(b * 3 + c) * HWSZ + yy * 512 + xx];
    }
  }
  __syncthreads();

  int j = threadIdx.y + 1, k = threadIdx.x + 1;  // center in 18-grid
  int hy = ty0 + threadIdx.y, hx = tx0 + threadIdx.x;
  for (int c = 0; c < 3; ++c) {
    float n2 = 0.25f *
        ( sG[0][c][j][k - 1]     - 2.0f * sG[0][c][j][k] + sG[0][c][j][k + 1]
        + sG[1][c][j - 1][k]     - 2.0f * sG[1][c][j][k] + sG[1][c][j + 1][k]
        + sG[2][c][j - 1][k - 1] - 2.0f * sG[2][c][j][k] + sG[2][c][j + 1][k + 1]
        + sG[3][c][j - 1][k + 1] - 2.0f * sG[3][c][j][k] + sG[3][c][j + 1][k - 1]);
    float n1 = -0.25f * (sD[c][j][k - 1] + sD[c][j][k + 1] + sD[c][j - 1][k] +
                         sD[c][j + 1][k] - 4.0f * sD[c][j][k]);
    float s0v = sS[c][j + 1][k + 1] - sD[c][j][k];
    rhs[(b * 3 + c) * HWSZ + hy * 512 + hx] = s0v + ALPHA_C * n1 + lam * n2;
  }
}

// ---------------- low = clip(S,0,1); high = imgs - low ----------------------
__global__ void __launch_bounds__(256)
k_finalize(const float* __restrict__ S, const float* __restrict__ img,
           float* __restrict__ out, int n) {
  int i = blockIdx.x * blockDim.x + threadIdx.x;
  if (i < n) {
    float lo = fminf(fmaxf(S[i], 0.0f), 1.0f);
    out[i] = lo;
    out[n + i] = img[i] - lo;
  }
}

extern "C" void kernel_launch(void* const* d_in, const int* in_sizes, int n_in,
                              void* d_out, int out_size, void* d_ws,
                              size_t ws_size, hipStream_t stream) {
  const float* imgs = (const float*)d_in[0];
  float* out = (float*)d_out;
  float* ws  = (float*)d_ws;
  const int NP = 4 * 3 * 512 * 512;  // one plane (12 MB)
  // Lifetime-aliased planes (48 MB total):
  float* P0 = ws;             // RHS  / T2 re
  float* P1 = ws + (size_t)NP;        // T2 im
  float* P2 = ws + 2 * (size_t)NP;    // T1 re / S
  float* P3 = ws + 3 * (size_t)NP;    // T1 im

  dim3 gsten(32, 32, 4), bsten(16, 16);
  const float* Scur = imgs;  // S starts as the input image
  float lam = 10.0f * BETA_C;
  for (int it = 0; it < 500 && lam <= 100.0f; ++it) {   // 14 iterations
    k_grad_rhs<<<gsten, bsten, 0, stream>>>(Scur, imgs, P0, lam);
    k_row_fwd<<<384, 512, 0, stream>>>(P0, P2, P3);
    k_col_solve<<<384, 512, 0, stream>>>(P2, P3, P0, P1, lam);
    k_row_inv<<<384, 512, 0, stream>>>(P0, P1, P2);
    Scur = P2;
    lam *= 1.5f;
  }
  k_finalize<<<(NP + 255) / 256, 256, 0, stream>>>(P2, imgs, out, NP);
}